// CoLightAgent_80564996538675
// MI455X (gfx1250) — compile-verified
//
#include <hip/hip_runtime.h>
#include <hip/hip_bf16.h>

// ---------------------------------------------------------------------------
// CoLight agent forward, restructured:
//   emb  = relu(relu(obs@We1+be1)@We2+be2)         (B=4, N=256, T=256)
//   ne   = adj @ emb[b]                             (per batch 256x256)
//   Sn   = sum_rows relu(ne@Wn+bn)                  (B, 256)  [column sums]
//   Sh   = sum_rows relu(ne@Wh+bh)                  (B, 256)
//   a    = relu(emb[b,tgt]@Wl+bl)  -> 8 softmaxes over 32 dims -> g (B,32)
//   act  = g @ Wa + ba                              (B, 8)
// GEMMs run on V_WMMA_F32_16X16X4_F32; each wave32 owns a 16x32 output strip
// (two accumulators) so the A fragment is fetched once per k-step and reused.
// ---------------------------------------------------------------------------

typedef __attribute__((ext_vector_type(2))) float v2f;
typedef __attribute__((ext_vector_type(8))) float v8f;

#define Bsz 4
#define Nn  256
#define OBS 40
#define ACT 8
#define HEAD 8
#define DIM 32
#define Tt  256   // HEAD*DIM

// Generic batched WMMA GEMM:  out = act(A @ W + bias), optional column-sum.
//   A: per-batch row-major (M x K), row stride lda, batch stride sA
//   W: per-batch row-major (K x Ncols), batch stride sW
//   out (colsum==0): per-batch row-major (M x Ncols), batch stride sO
//   out (colsum==1): per-batch length-Ncols accumulator (atomic add of
//                    column sums of the activated strip), batch stride sO
// grid: (Ncols/32, M/16, batches), block: 32 (one wave; EXEC all ones).
__global__ __launch_bounds__(32)
void wmma_gemm_f32(const float* __restrict__ A, long long sA, int lda,
                   const float* __restrict__ W, long long sW, int ldw,
                   const float* __restrict__ bias,
                   float* __restrict__ out, long long sO, int ldo,
                   int K, int do_relu, int colsum)
{
    const int lane   = threadIdx.x;          // 0..31
    const int lane16 = lane & 15;
    const int khi    = (lane >> 4) << 1;     // 0 for lanes 0-15, 2 for 16-31
    const int b      = blockIdx.z;

    const float* Wb = W + (long long)b * sW;

    const int rowA = blockIdx.y * 16 + lane16;   // A-matrix row for this lane
    const int col0 = blockIdx.x * 32 + lane16;   // first column tile
    const int col1 = col0 + 16;                  // second column tile

    v8f acc0 = {0.f, 0.f, 0.f, 0.f, 0.f, 0.f, 0.f, 0.f};
    v8f acc1 = {0.f, 0.f, 0.f, 0.f, 0.f, 0.f, 0.f, 0.f};

    const float* aRow = A + (long long)b * sA + (long long)rowA * lda;
    for (int k0 = 0; k0 < K; k0 += 4) {
        const int ka = k0 + khi;
        v2f afrag, bfrag0, bfrag1;
        afrag.x = aRow[ka];                      // compiler merges -> b64 load
        afrag.y = aRow[ka + 1];
        const float* wr0 = Wb + (long long)ka * ldw;
        const float* wr1 = wr0 + ldw;
        bfrag0.x = wr0[col0];
        bfrag0.y = wr1[col0];
        bfrag1.x = wr0[col1];
        bfrag1.y = wr1[col1];
        // D = A(16x4) * B(4x16) + C in full fp32; A fragment reused twice
        acc0 = __builtin_amdgcn_wmma_f32_16x16x4_f32(
                   false, afrag, false, bfrag0, (short)0, acc0, false, false);
        acc1 = __builtin_amdgcn_wmma_f32_16x16x4_f32(
                   false, afrag, false, bfrag1, (short)0, acc1, false, false);
    }

    const float bc0 = bias ? bias[col0] : 0.f;
    const float bc1 = bias ? bias[col1] : 0.f;

    if (!colsum) {
        float* Ob = out + (long long)b * sO;
        const int rowBase = blockIdx.y * 16 + ((lane >> 4) << 3); // +8 upper half
        #pragma unroll
        for (int r = 0; r < 8; ++r) {
            float v0 = acc0[r] + bc0;
            float v1 = acc1[r] + bc1;
            if (do_relu) {
                v0 = v0 > 0.f ? v0 : 0.f;
                v1 = v1 > 0.f ? v1 : 0.f;
            }
            Ob[(long long)(rowBase + r) * ldo + col0] = v0;
            Ob[(long long)(rowBase + r) * ldo + col1] = v1;
        }
    } else {
        float p0 = 0.f, p1 = 0.f;
        #pragma unroll
        for (int r = 0; r < 8; ++r) {
            float v0 = acc0[r] + bc0;
            float v1 = acc1[r] + bc1;
            if (do_relu) {
                v0 = v0 > 0.f ? v0 : 0.f;
                v1 = v1 > 0.f ? v1 : 0.f;
            }
            p0 += v0;
            p1 += v1;
        }
        float* Sb = out + (long long)b * sO;
        atomicAdd(Sb + col0, p0);
        atomicAdd(Sb + col1, p1);
    }
}

__global__ void zero_f32(float* p, int n)
{
    int i = blockIdx.x * blockDim.x + threadIdx.x;
    if (i < n) p[i] = 0.f;
}

// Final combine: ah row at tgt[b], per-head softmax over DIM, mix with Sh,
// mean over heads, project with Wa. One block per batch, 256 threads.
__global__ __launch_bounds__(256)
void final_combine(const float* __restrict__ x,
                   const float* __restrict__ emb,
                   const float* __restrict__ Sn,
                   const float* __restrict__ Sh,
                   const float* __restrict__ Wl, const float* __restrict__ bl,
                   const float* __restrict__ Wa, const float* __restrict__ ba,
                   float* __restrict__ act)
{
    const int b = blockIdx.x;
    const int t = threadIdx.x;   // 0..255

    __shared__ float a_sh[Tt];
    __shared__ float gpart[HEAD][DIM];
    __shared__ float g[DIM];

    const int tgt = (int)x[((long long)b * (Nn + 1) + Nn) * OBS];  // x[b,-1,0]
    const float* e = emb + ((long long)b * Nn + tgt) * Tt;

    // a[t] = relu(emb[b,tgt] . Wl[:,t] + bl[t])
    float acc = bl[t];
    for (int k = 0; k < Tt; ++k) acc += e[k] * Wl[k * Tt + t];
    a_sh[t] = acc > 0.f ? acc : 0.f;
    __syncthreads();

    // per-head softmax over DIM, weight by Sh column sums
    if (t < HEAD) {
        const int h = t;
        const float* SnB = Sn + b * Tt;
        const float* ShB = Sh + b * Tt;
        float logits[DIM];
        float m = -3.4e38f;
        for (int d = 0; d < DIM; ++d) {
            float l = a_sh[d * HEAD + h] * SnB[d * HEAD + h];
            logits[d] = l;
            m = l > m ? l : m;
        }
        float s = 0.f;
        for (int d = 0; d < DIM; ++d) {
            float p = __expf(logits[d] - m);
            logits[d] = p;
            s += p;
        }
        const float inv = 1.f / s;
        for (int d = 0; d < DIM; ++d)
            gpart[h][d] = logits[d] * inv * ShB[d * HEAD + h];
    }
    __syncthreads();

    if (t < DIM) {
        float s = 0.f;
        for (int h = 0; h < HEAD; ++h) s += gpart[h][t];
        g[t] = s * (1.f / HEAD);
    }
    __syncthreads();

    if (t < ACT) {
        float s = ba[t];
        for (int d = 0; d < DIM; ++d) s += g[d] * Wa[d * ACT + t];
        act[b * ACT + t] = s;
    }
}

extern "C" void kernel_launch(void* const* d_in, const int* in_sizes, int n_in,
                              void* d_out, int out_size, void* d_ws, size_t ws_size,
                              hipStream_t stream)
{
    (void)in_sizes; (void)n_in; (void)out_size; (void)ws_size;

    const float* x   = (const float*)d_in[0];
    const float* adj = (const float*)d_in[1];
    const float* We1 = (const float*)d_in[2];
    const float* be1 = (const float*)d_in[3];
    const float* We2 = (const float*)d_in[4];
    const float* be2 = (const float*)d_in[5];
    const float* Wl  = (const float*)d_in[6];
    const float* bl  = (const float*)d_in[7];
    const float* Wn  = (const float*)d_in[8];
    const float* bn  = (const float*)d_in[9];
    const float* Wh  = (const float*)d_in[10];
    const float* bh  = (const float*)d_in[11];
    const float* Wa  = (const float*)d_in[12];
    const float* ba  = (const float*)d_in[13];
    float* out = (float*)d_out;

    // workspace layout (floats)
    float* ws   = (float*)d_ws;
    float* emb1 = ws;                       // 4*256*256 = 262144
    float* emb  = emb1 + Bsz * Nn * Tt;     // 262144
    float* ne   = emb  + Bsz * Nn * Tt;     // 262144
    float* Sn   = ne   + Bsz * Nn * Tt;     // 1024
    float* Sh   = Sn   + Bsz * Tt;          // 1024

    const long long sBatchEmb = (long long)Nn * Tt;        // 65536
    const long long sBatchX   = (long long)(Nn + 1) * OBS; // 10280 (skip last row)

    // zero the column-sum accumulators
    zero_f32<<<dim3((2 * Bsz * Tt + 255) / 256), dim3(256), 0, stream>>>(Sn, 2 * Bsz * Tt);

    dim3 blk(32);
    dim3 grd(Tt / 32, Nn / 16, Bsz);   // 16x32 strip per wave

    // emb1 = relu(obs @ We1 + be1)   (K = 40)
    wmma_gemm_f32<<<grd, blk, 0, stream>>>(x, sBatchX, OBS,
                                           We1, 0, Tt, be1,
                                           emb1, sBatchEmb, Tt,
                                           OBS, 1, 0);
    // emb = relu(emb1 @ We2 + be2)   (K = 256)
    wmma_gemm_f32<<<grd, blk, 0, stream>>>(emb1, sBatchEmb, Tt,
                                           We2, 0, Tt, be2,
                                           emb, sBatchEmb, Tt,
                                           Tt, 1, 0);
    // ne[b] = adj @ emb[b]           (K = 256, shared A, batched B)
    wmma_gemm_f32<<<grd, blk, 0, stream>>>(adj, 0, Nn,
                                           emb, sBatchEmb, Tt, nullptr,
                                           ne, sBatchEmb, Tt,
                                           Nn, 0, 0);
    // Sn[b] = colsum relu(ne[b] @ Wn + bn)
    wmma_gemm_f32<<<grd, blk, 0, stream>>>(ne, sBatchEmb, Tt,
                                           Wn, 0, Tt, bn,
                                           Sn, Tt, Tt,
                                           Tt, 1, 1);
    // Sh[b] = colsum relu(ne[b] @ Wh + bh)
    wmma_gemm_f32<<<grd, blk, 0, stream>>>(ne, sBatchEmb, Tt,
                                           Wh, 0, Tt, bh,
                                           Sh, Tt, Tt,
                                           Tt, 1, 1);
    // final gather + softmax + projection
    final_combine<<<dim3(Bsz), dim3(256), 0, stream>>>(x, emb, Sn, Sh,
                                                       Wl, bl, Wa, ba, out);
}